// TGCN_RecurrentGCN_45801531244830
// MI455X (gfx1250) — compile-verified
//
#include <hip/hip_runtime.h>
#include <hip/hip_bf16.h>
#include <cstdint>

// ---------------------------------------------------------------------------
// TGCN recurrent cell for MI455X (gfx1250).
// GEMMs via V_WMMA_F32_16X16X4_F32 (full f32 precision; GEMM is not the
// bottleneck -- edge aggregation is, and it is L2-resident on a 192MB L2).
// Edge streams use non-temporal loads so they don't evict xw/agg from L2.
// ---------------------------------------------------------------------------

typedef float v2f __attribute__((ext_vector_type(2)));
typedef float v8f __attribute__((ext_vector_type(8)));

__device__ __forceinline__ v8f wmma_f32_4(v2f a, v2f b, v8f c) {
  // 8 args: (neg_a, A, neg_b, B, c_mod, C, reuse_a, reuse_b)
  return __builtin_amdgcn_wmma_f32_16x16x4_f32(false, a, false, b, (short)0, c,
                                               false, false);
}

#define NF 128   // input feature dim
#define NH 32    // hidden dim
#define NG 96    // 3 * NH (fused Wz|Wr|Wh output)
#define NC 10    // classes

// ---- deg init: every node gets self-loop weight 1 --------------------------
__global__ void k_initdeg(float* __restrict__ deg, int N) {
  int i = blockIdx.x * blockDim.x + threadIdx.x;
  if (i < N) deg[i] = 1.0f;
}

// ---- deg += edge weights scattered by dst (NT edge stream) -----------------
__global__ void k_deg(const long long* __restrict__ ei,
                      const float* __restrict__ ew,
                      float* __restrict__ deg, long long E) {
  long long e = (long long)blockIdx.x * blockDim.x + threadIdx.x;
  if (e < E) {
    long long d = __builtin_nontemporal_load(ei + E + e);
    float     w = __builtin_nontemporal_load(ew + e);
    atomicAdd(&deg[d], w);
  }
}

// ---- XW = x[N,128] @ [Wz|Wr|Wh][128,96] via WMMA f32 16x16x4 ---------------
// block = 128 threads (4 waves). Block covers 64 rows; each wave owns 16.
// x tile in LDS stride 132 (132 % 64 == 4 -> A-frag rows on distinct banks for
// both half-waves). W staged per gate in LDS stride 40 (2*40 % 64 == 16 ->
// half-waves hit disjoint bank groups). Inner loop: ds_load + v_wmma only.
__global__ __launch_bounds__(128) void k_xw(const float* __restrict__ x,
                                            const float* __restrict__ Wz,
                                            const float* __restrict__ Wr,
                                            const float* __restrict__ Wh,
                                            float* __restrict__ xw, int N) {
  __shared__ float xs[64][132];     // 33.8 KB
  __shared__ float Wst[128][40];    // 20.5 KB
  const int t = threadIdx.x;
  const int rowbase = blockIdx.x * 64;
  for (int i = 0; i < 64; ++i) {
    int r = rowbase + i;
    int rc = r < N ? r : N - 1;     // clamp; garbage rows never stored
    xs[i][t] = x[(size_t)rc * NF + t];
  }
  const int wv = t >> 5, lane = t & 31;
  const int m = lane & 15, hi = lane >> 4, koff = hi * 2;

#pragma unroll
  for (int g = 0; g < 3; ++g) {
    const float* W = (g == 0) ? Wz : (g == 1) ? Wr : Wh;  // literal per unroll
    __syncthreads();                 // prior-gate reads done (and xs visible)
    for (int idx = t; idx < NF * NH; idx += 128)
      Wst[idx >> 5][idx & 31] = W[idx];
    __syncthreads();
    for (int nt2 = 0; nt2 < 2; ++nt2) {
      const int nloc = nt2 * 16 + m;
      v8f acc = {};
      for (int kb = 0; kb < NF; kb += 4) {
        const int k0 = kb + koff;
        v2f a; a.x = xs[wv * 16 + m][k0];  a.y = xs[wv * 16 + m][k0 + 1];
        v2f b; b.x = Wst[k0][nloc];        b.y = Wst[k0 + 1][nloc];
        acc = wmma_f32_4(a, b, acc);
      }
      for (int r = 0; r < 8; ++r) {  // D: row = r + 8*hi, col = m
        int row = rowbase + wv * 16 + r + hi * 8;
        if (row < N) xw[(size_t)row * NG + g * NH + nloc] = acc[r];
      }
    }
  }
}

// ---- dinv = rsqrt(deg); agg seeded with self-loop message xw*dinv^2 --------
__global__ void k_selfloop(const float* __restrict__ deg,
                           const float* __restrict__ xw,
                           float* __restrict__ dinv,
                           float* __restrict__ agg, int N) {
  int i = blockIdx.x * blockDim.x + threadIdx.x;
  if (i < N * NG) {
    int n = i / NG, c = i - n * NG;
    float d = deg[n];
    float di = d > 0.f ? rsqrtf(d) : 0.f;
    if (c == 0) dinv[n] = di;
    agg[i] = xw[i] * di * di;
  }
}

// ---- edge scatter: agg[dst] += xw[src] * (dinv[src]*w*dinv[dst]) -----------
// One wave per edge; lanes cover the 96 columns (3 passes). xw and agg are
// L2-resident (38.4MB each); edge stream is NT so it doesn't evict them.
__global__ void k_scatter(const long long* __restrict__ ei,
                          const float* __restrict__ ew,
                          const float* __restrict__ xw,
                          const float* __restrict__ dinv,
                          float* __restrict__ agg, long long E) {
  const int lane = threadIdx.x & 31;
  long long wid = (long long)blockIdx.x * (blockDim.x >> 5) + (threadIdx.x >> 5);
  long long nw  = (long long)gridDim.x * (blockDim.x >> 5);
  for (long long e = wid; e < E; e += nw) {
    long long s  = __builtin_nontemporal_load(ei + e);
    long long d  = __builtin_nontemporal_load(ei + E + e);
    float     wt = __builtin_nontemporal_load(ew + e);
    float nrm = dinv[s] * wt * dinv[d];
    const float* xs_ = xw + (size_t)s * NG;
    float*       ad  = agg + (size_t)d * NG;
    for (int c = lane; c < NG; c += 32)
      atomicAdd(&ad[c], xs_[c] * nrm);
  }
}

// ---- fused GRU + head ------------------------------------------------------
// block = 128 threads (4 waves); wave owns 16 nodes. LDS tile Vs[node][128]
// holds [Az'|Ar'|Ah'|H] (Ag' = agg + gcn bias). Gate GEMMs ([16,64]@[64,32])
// run as WMMA with K-column remapping; HR recycles cols 0..31, relu(H_new)
// recycles cols 32..63. Head (32x10 + softmax) is scalar: 64 FLOP/output.
__global__ __launch_bounds__(128) void k_gru(
    const float* __restrict__ agg, const float* __restrict__ H,
    const float* __restrict__ bz, const float* __restrict__ br,
    const float* __restrict__ bh,
    const float* __restrict__ Lz, const float* __restrict__ Lzb,
    const float* __restrict__ Lr, const float* __restrict__ Lrb,
    const float* __restrict__ Lh, const float* __restrict__ Lhb,
    const float* __restrict__ lin_w, const float* __restrict__ lin_b,
    float* __restrict__ outP, float* __restrict__ outH, int N) {
  __shared__ float Vs[64][132];      // 33.8 KB
  __shared__ float Lws[3][64][40];   // 30.7 KB, stride 40: 2*40 % 64 == 16 ->
                                     // half-waves hit disjoint bank groups
  const int t = threadIdx.x;
  const int nodebase = blockIdx.x * 64;

  // stage Vs: warp-uniform branches, concrete pointers, bias hoisted
  if (t < 96) {
    float addv = (t < 32) ? bz[t] : (t < 64) ? br[t - 32] : bh[t - 64];
    for (int i = 0; i < 64; ++i) {
      int r = nodebase + i;
      int rc = r < N ? r : N - 1;
      Vs[i][t] = agg[(size_t)rc * NG + t] + addv;
    }
  } else {
    int c = t - 96;
    for (int i = 0; i < 64; ++i) {
      int r = nodebase + i;
      int rc = r < N ? r : N - 1;
      Vs[i][96 + c] = H[(size_t)rc * NH + c];
    }
  }
  // stage gate weights: three literal-pointer loops -> global_load
  for (int idx = t; idx < 2048; idx += 128) Lws[0][idx >> 5][idx & 31] = Lz[idx];
  for (int idx = t; idx < 2048; idx += 128) Lws[1][idx >> 5][idx & 31] = Lr[idx];
  for (int idx = t; idx < 2048; idx += 128) Lws[2][idx >> 5][idx & 31] = Lh[idx];
  __syncthreads();

  const int wv = t >> 5, lane = t & 31;
  const int m = lane & 15, hi = lane >> 4, koff = hi * 2;
  const int rowL = wv * 16;

  v8f Zacc[2], Racc[2], Tacc[2];
  // Z gate: K<32 -> Az' (cols 0..31), K>=32 -> H (cols 96..127)
  for (int tt = 0; tt < 2; ++tt) {
    v8f acc = {};
    const int n = tt * 16 + m;
    for (int kb = 0; kb < 64; kb += 4) {
      const int k0 = kb + koff;
      const int c0 = (k0 < 32) ? k0 : (96 + k0 - 32);
      v2f a; a.x = Vs[rowL + m][c0];   a.y = Vs[rowL + m][c0 + 1];
      v2f b; b.x = Lws[0][k0][n];      b.y = Lws[0][k0 + 1][n];
      acc = wmma_f32_4(a, b, acc);
    }
    Zacc[tt] = acc;
  }
  // R gate: K<32 -> Ar' (cols 32..63)
  for (int tt = 0; tt < 2; ++tt) {
    v8f acc = {};
    const int n = tt * 16 + m;
    for (int kb = 0; kb < 64; kb += 4) {
      const int k0 = kb + koff;
      const int c0 = (k0 < 32) ? (32 + k0) : (96 + k0 - 32);
      v2f a; a.x = Vs[rowL + m][c0];   a.y = Vs[rowL + m][c0 + 1];
      v2f b; b.x = Lws[1][k0][n];      b.y = Lws[1][k0 + 1][n];
      acc = wmma_f32_4(a, b, acc);
    }
    Racc[tt] = acc;
  }
  // sigmoid; HR = H * R written into freed cols 0..31
  for (int tt = 0; tt < 2; ++tt) {
    const int n = tt * 16 + m;
    for (int r = 0; r < 8; ++r) {
      const int mr = r + hi * 8;
      float z  = 1.f / (1.f + __expf(-(Zacc[tt][r] + Lzb[n])));
      float rv = 1.f / (1.f + __expf(-(Racc[tt][r] + Lrb[n])));
      Zacc[tt][r] = z;
      Vs[rowL + mr][n] = Vs[rowL + mr][96 + n] * rv;
    }
  }
  __syncthreads();
  // H_tilde: K<32 -> Ah' (cols 64..95), K>=32 -> HR (cols 0..31)
  for (int tt = 0; tt < 2; ++tt) {
    v8f acc = {};
    const int n = tt * 16 + m;
    for (int kb = 0; kb < 64; kb += 4) {
      const int k0 = kb + koff;
      const int c0 = (k0 < 32) ? (64 + k0) : (k0 - 32);
      v2f a; a.x = Vs[rowL + m][c0];   a.y = Vs[rowL + m][c0 + 1];
      v2f b; b.x = Lws[2][k0][n];      b.y = Lws[2][k0 + 1][n];
      acc = wmma_f32_4(a, b, acc);
    }
    Tacc[tt] = acc;
  }
  // H_new = Z*H + (1-Z)*tanh(.); relu -> cols 32..63
  for (int tt = 0; tt < 2; ++tt) {
    const int n = tt * 16 + m;
    for (int r = 0; r < 8; ++r) {
      const int mr = r + hi * 8;
      float ht = tanhf(Tacc[tt][r] + Lhb[n]);
      float z  = Zacc[tt][r];
      float hv = Vs[rowL + mr][96 + n];
      float hn = z * hv + (1.f - z) * ht;
      Vs[rowL + mr][32 + n] = hn > 0.f ? hn : 0.f;
    }
  }
  __syncthreads();

  // output #2: copy of input H (reference returns original H)
  for (int idx = lane; idx < 16 * NH; idx += 32) {
    int nn = idx >> 5, c = idx & 31;
    int node = nodebase + rowL + nn;
    if (node < N) outH[(size_t)node * NH + c] = Vs[rowL + nn][96 + c];
  }
  // output #1: softmax(relu(H_new) @ lin_w + lin_b); lanes 0..15, node each
  if (hi == 0) {
    int node = nodebase + rowL + m;
    if (node < N) {
      float lg[NC], mx = -1e30f;
      for (int c = 0; c < NC; ++c) {
        float s = lin_b[c];
        for (int k = 0; k < NH; ++k) s += Vs[rowL + m][32 + k] * lin_w[k * NC + c];
        lg[c] = s; mx = fmaxf(mx, s);
      }
      float se = 0.f;
      for (int c = 0; c < NC; ++c) { lg[c] = __expf(lg[c] - mx); se += lg[c]; }
      float inv = 1.f / se;
      for (int c = 0; c < NC; ++c) outP[(size_t)node * NC + c] = lg[c] * inv;
    }
  }
}

// ---------------------------------------------------------------------------
extern "C" void kernel_launch(void* const* d_in, const int* in_sizes, int n_in,
                              void* d_out, int out_size, void* d_ws, size_t ws_size,
                              hipStream_t stream) {
  const float*     x   = (const float*)d_in[0];
  const long long* ei  = (const long long*)d_in[1];   // int64 [2,E]
  const float*     ew  = (const float*)d_in[2];
  const float*     H   = (const float*)d_in[3];
  const float*     Wz  = (const float*)d_in[4];
  const float*     bz  = (const float*)d_in[5];
  const float*     Wr  = (const float*)d_in[6];
  const float*     br  = (const float*)d_in[7];
  const float*     Wh  = (const float*)d_in[8];
  const float*     bh  = (const float*)d_in[9];
  const float*     Lz  = (const float*)d_in[10];
  const float*     Lzb = (const float*)d_in[11];
  const float*     Lr  = (const float*)d_in[12];
  const float*     Lrb = (const float*)d_in[13];
  const float*     Lh  = (const float*)d_in[14];
  const float*     Lhb = (const float*)d_in[15];
  const float*     lw  = (const float*)d_in[16];
  const float*     lb  = (const float*)d_in[17];

  const int       N = in_sizes[0] / NF;
  const long long E = in_sizes[2];

  float* xw   = (float*)d_ws;            // N*96
  float* agg  = xw  + (size_t)N * NG;    // N*96
  float* deg  = agg + (size_t)N * NG;    // N
  float* dinv = deg + N;                 // N

  float* outP = (float*)d_out;           // N*10
  float* outH = outP + (size_t)N * NC;   // N*32

  k_initdeg<<<(N + 255) / 256, 256, 0, stream>>>(deg, N);
  k_deg<<<(int)((E + 255) / 256), 256, 0, stream>>>(ei, ew, deg, E);
  k_xw<<<(N + 63) / 64, 128, 0, stream>>>(x, Wz, Wr, Wh, xw, N);
  k_selfloop<<<(N * NG + 255) / 256, 256, 0, stream>>>(deg, xw, dinv, agg, N);
  k_scatter<<<2048, 256, 0, stream>>>(ei, ew, xw, dinv, agg, E);
  k_gru<<<(N + 63) / 64, 128, 0, stream>>>(agg, H, bz, br, bh,
                                           Lz, Lzb, Lr, Lrb, Lh, Lhb,
                                           lw, lb, outP, outH, N);
}